// RetrievalModule_89000312308159
// MI455X (gfx1250) — compile-verified
//
#include <hip/hip_runtime.h>
#include <math.h>

#define D        256
#define TOPK     32
#define NROWS    2048            // 4 * 512 query rows, flattened
#define NKEYS    65536
#define NCHUNK   16              // key chunks (phase-1 grid.x)
#define CHUNK_KEYS (NKEYS / NCHUNK)   // 4096 keys per chunk
#define CAND_PER_ROW (NCHUNK * 64)    // 16 chunks * 2 lanes * 32 = 1024
#define SCALE    0.0625f         // 256^-0.5

typedef float v2f __attribute__((ext_vector_type(2)));
typedef float v8f __attribute__((ext_vector_type(8)));

// ---------------------------------------------------------------------------
// Phase 1: WMMA f32 GEMM (Q tile x key chunk) + per-lane running top-32.
// One wave32 per workgroup. Lane L: half = L>>4, r = L&15.
//   A (16x4 f32): lanes 0-15 rows M=lane K={0,1}; lanes 16-31 K={2,3}
//   B (4x16 f32): same pattern with N=lane&15  -> both are 8B loads at
//   row*D + 4*s + 2*half, so Q and K fragments use identical addressing.
//   C (16x16 f32): VGPR i -> M = i + 8*half, N = lane&15.
// ---------------------------------------------------------------------------
__global__ __launch_bounds__(32)
void retr_phase1(const float* __restrict__ Q, const float* __restrict__ K,
                 float* __restrict__ wsS, int* __restrict__ wsI)
{
    const int lane  = threadIdx.x;
    const int half  = lane >> 4;
    const int r     = lane & 15;
    const int qtile = blockIdx.y;            // 0..127
    const int chunk = blockIdx.x;            // 0..15
    const int row_g = qtile * 16 + r;        // global query row for this lane

    // Preload all A fragments for this query tile (64 K-steps, reused 256x).
    v2f a[64];
    const float* qrow = Q + (size_t)row_g * D + 2 * half;
    #pragma unroll
    for (int s = 0; s < 64; ++s)
        a[s] = *(const v2f*)(qrow + 4 * s);

    // Per-lane running top-32 (this lane covers a disjoint half of the keys
    // for row r: cols 0-7 of every tile if half==0, cols 8-15 if half==1).
    float tv[TOPK]; int ti[TOPK];
    #pragma unroll
    for (int j = 0; j < TOPK; ++j) { tv[j] = -INFINITY; ti[j] = 0; }
    float curmin = -INFINITY; int minpos = 0;

    __shared__ float stile[16][17];          // padded to dodge bank conflicts

    const int kbase = chunk * CHUNK_KEYS;
    for (int t = 0; t < CHUNK_KEYS / 16; ++t) {
        const int n0 = kbase + t * 16;
        const float* krow = K + (size_t)(n0 + r) * D + 2 * half;
        if (t + 1 < CHUNK_KEYS / 16)
            __builtin_prefetch(krow + 16 * D, 0, 0);   // next key tile

        v8f c = {0.f, 0.f, 0.f, 0.f, 0.f, 0.f, 0.f, 0.f};
        #pragma unroll
        for (int s = 0; s < 64; ++s) {
            v2f b = *(const v2f*)(krow + 4 * s);
            c = __builtin_amdgcn_wmma_f32_16x16x4_f32(
                    /*neg_a=*/false, a[s], /*neg_b=*/false, b,
                    /*c_mod=*/(short)0, c, /*reuse_a=*/false, /*reuse_b=*/false);
        }

        // Transpose C layout through LDS: lane writes rows i+8*half, col r.
        #pragma unroll
        for (int i = 0; i < 8; ++i)
            stile[i + 8 * half][r] = c[i];
        __syncthreads();

        // Scan: lane handles row r, key columns half*8 .. half*8+7.
        #pragma unroll
        for (int j = 0; j < 8; ++j) {
            const int   col = half * 8 + j;
            const float sc  = stile[r][col];
            if (sc > curmin) {
                const int idx = n0 + col;
                #pragma unroll
                for (int u = 0; u < TOPK; ++u)
                    if (u == minpos) { tv[u] = sc; ti[u] = idx; }
                curmin = tv[0]; minpos = 0;
                #pragma unroll
                for (int u = 1; u < TOPK; ++u)
                    if (tv[u] < curmin) { curmin = tv[u]; minpos = u; }
            }
        }
        __syncthreads();
    }

    // Emit 32 candidates; 2 lanes/row -> 64 per (row, chunk).
    const size_t base = ((size_t)row_g * NCHUNK + chunk) * 64 + half * 32;
    #pragma unroll
    for (int j = 0; j < TOPK; ++j) { wsS[base + j] = tv[j]; wsI[base + j] = ti[j]; }
}

// ---------------------------------------------------------------------------
// Phase 2: one wave per query row. Exact top-32 of 1024 candidates via 32
// rounds of register-scan + shfl_xor argmax, then softmax + value gather.
// ---------------------------------------------------------------------------
__global__ __launch_bounds__(256)
void retr_phase2(const float* __restrict__ wsS, const int* __restrict__ wsI,
                 const float* __restrict__ V, float* __restrict__ out)
{
    const int lane = threadIdx.x & 31;
    const int wave = threadIdx.x >> 5;
    const int row  = blockIdx.x * 8 + wave;

    // 32 candidates per lane, coalesced strided load.
    float cv[32]; int ci[32];
    const size_t base = (size_t)row * CAND_PER_ROW;
    #pragma unroll
    for (int j = 0; j < 32; ++j) {
        cv[j] = wsS[base + lane + 32 * j];
        ci[j] = wsI[base + lane + 32 * j];
    }

    float selv[TOPK]; int seli[TOPK];
    #pragma unroll
    for (int it = 0; it < TOPK; ++it) {
        // Local argmax over this lane's 32 registers.
        float bv = cv[0]; int bj = 0;
        #pragma unroll
        for (int j = 1; j < 32; ++j)
            if (cv[j] > bv) { bv = cv[j]; bj = j; }
        int bidx = ci[bj], bslot = bj, blane = lane;

        // Cross-lane max (tie-break on smaller key index).
        #pragma unroll
        for (int m = 16; m >= 1; m >>= 1) {
            float ov    = __shfl_xor(bv, m, 32);
            int   oidx  = __shfl_xor(bidx, m, 32);
            int   oslot = __shfl_xor(bslot, m, 32);
            int   olane = __shfl_xor(blane, m, 32);
            if (ov > bv || (ov == bv && oidx < bidx)) {
                bv = ov; bidx = oidx; bslot = oslot; blane = olane;
            }
        }
        selv[it] = bv; seli[it] = bidx;          // uniform across the wave
        if (lane == blane) {                      // retire the winner
            #pragma unroll
            for (int j = 0; j < 32; ++j)
                if (j == bslot) cv[j] = -INFINITY;
        }
    }

    // Softmax over the 32 selected (selv[0] is the max). Apply score scale here
    // (monotonic, so deferring it past top-k is exact).
    const float mx = selv[0];
    float sum = 0.f;
    #pragma unroll
    for (int k = 0; k < TOPK; ++k) {
        selv[k] = expf((selv[k] - mx) * SCALE);   // reuse selv as weights
        sum += selv[k];
    }
    const float inv = 1.0f / sum;

    // Weighted gather: lane owns 8 of 256 dims.
    const int d0 = lane * 8;
    float acc[8];
    #pragma unroll
    for (int i = 0; i < 8; ++i) acc[i] = 0.f;
    #pragma unroll
    for (int k = 0; k < TOPK; ++k) {
        const float* vp = V + (size_t)seli[k] * D + d0;
        const float4 x = *(const float4*)vp;
        const float4 y = *(const float4*)(vp + 4);
        const float wk = selv[k] * inv;
        acc[0] += wk * x.x; acc[1] += wk * x.y; acc[2] += wk * x.z; acc[3] += wk * x.w;
        acc[4] += wk * y.x; acc[5] += wk * y.y; acc[6] += wk * y.z; acc[7] += wk * y.w;
    }
    float* op = out + (size_t)row * D + d0;
    *(float4*)op       = make_float4(acc[0], acc[1], acc[2], acc[3]);
    *(float4*)(op + 4) = make_float4(acc[4], acc[5], acc[6], acc[7]);
}

// ---------------------------------------------------------------------------
extern "C" void kernel_launch(void* const* d_in, const int* in_sizes, int n_in,
                              void* d_out, int out_size, void* d_ws, size_t ws_size,
                              hipStream_t stream)
{
    const float* Q = (const float*)d_in[0];   // [4,512,256]
    const float* K = (const float*)d_in[1];   // [65536,256]
    const float* V = (const float*)d_in[2];   // [65536,256]
    float* out = (float*)d_out;               // [4,512,256]

    float* wsS = (float*)d_ws;                                        // 8 MB
    int*   wsI = (int*)((char*)d_ws +
                        (size_t)NROWS * CAND_PER_ROW * sizeof(float)); // 8 MB

    dim3 g1(NCHUNK, NROWS / 16);   // 16 x 128 workgroups, 1 wave each
    retr_phase1<<<g1, 32, 0, stream>>>(Q, K, wsS, wsI);

    retr_phase2<<<NROWS / 8, 256, 0, stream>>>(wsS, wsI, V, out);
}